// PatchSupCLRLoss_36584531427974
// MI455X (gfx1250) — compile-verified
//
#include <hip/hip_runtime.h>
#include <math.h>

// Problem constants (from the reference)
#define N_TOTAL   8192
#define NCLS      100
#define INV_T     10.0f            // 1 / 0.1
#define LOSS_SCALE (0.1f / 0.07f)  // T / BASE_T

typedef __attribute__((ext_vector_type(2))) float v2f;
typedef __attribute__((ext_vector_type(8))) float v8f;

// ---------------------------------------------------------------------------
// Kernel 1: deterministic per-class counting sort of (q*invT, k) + k min/max.
// Single block, 128 threads. Work is O(100 * 8192) label scans: trivial.
// ---------------------------------------------------------------------------
__global__ void supclr_prep_kernel(const float* __restrict__ q,
                                   const float* __restrict__ k,
                                   const int*   __restrict__ labels,
                                   float* __restrict__ q_sorted,
                                   float* __restrict__ k_sorted,
                                   int*   __restrict__ offsets,
                                   float* __restrict__ kminmax) {
  __shared__ int   s_counts[NCLS];
  __shared__ int   s_off[NCLS + 1];
  __shared__ float s_max[128];
  __shared__ float s_min[128];
  const int t = threadIdx.x;

  // per-class counts (stable, no atomics)
  if (t < NCLS) {
    int cnt = 0;
    for (int j = 0; j < N_TOTAL; ++j) cnt += (labels[j] == t);
    s_counts[t] = cnt;
  }

  // global k min / max (deterministic tree)
  float mx = -INFINITY, mn = INFINITY;
  for (int j = t; j < N_TOTAL; j += 128) {
    float v = k[j];
    mx = fmaxf(mx, v);
    mn = fminf(mn, v);
  }
  s_max[t] = mx;
  s_min[t] = mn;
  __syncthreads();

  if (t == 0) {
    int acc = 0;
    for (int c = 0; c < NCLS; ++c) { s_off[c] = acc; acc += s_counts[c]; }
    s_off[NCLS] = acc;
    for (int c = 0; c <= NCLS; ++c) offsets[c] = s_off[c];
    float gmx = -INFINITY, gmn = INFINITY;
    for (int i = 0; i < 128; ++i) {
      gmx = fmaxf(gmx, s_max[i]);
      gmn = fminf(gmn, s_min[i]);
    }
    kminmax[0] = gmn;
    kminmax[1] = gmx;
  }
  __syncthreads();

  // stable per-class compaction (one thread per class -> deterministic order)
  if (t < NCLS) {
    int pos = s_off[t];
    for (int j = 0; j < N_TOTAL; ++j) {
      if (labels[j] == t) {
        q_sorted[pos] = q[j] * INV_T;  // fold 1/T into q
        k_sorted[pos] = k[j];
        ++pos;
      }
    }
  }
}

// ---------------------------------------------------------------------------
// Kernel 2: per-class outer-product tiles via V_WMMA_F32_16X16X4_F32.
// One wave (32 threads) owns a 16-row tile of one class; loops over the
// class's 16-column tiles. A column K=0 carries q/T, B row K=0 carries k;
// D[m,n] = (q_m/T)*k_n. exp + mask + row-sum done on the accumulator regs.
// ---------------------------------------------------------------------------
__global__ void supclr_tile_kernel(const float* __restrict__ q_sorted,
                                   const float* __restrict__ k_sorted,
                                   const int*   __restrict__ offsets,
                                   const float* __restrict__ kminmax,
                                   float* __restrict__ loss_row) {
  const int c    = blockIdx.x;
  const int off  = offsets[c];
  const int n_c  = offsets[c + 1] - off;
  const int row0 = blockIdx.y * 16;
  if (row0 >= n_c) return;  // block-uniform: EXEC stays all-1s for WMMA

  const int  lane  = threadIdx.x;
  const bool lo    = (lane < 16);
  const int  nlane = lane & 15;

  const float kmin = kminmax[0];
  const float kmax = kminmax[1];

  // A fragment (16x4 f32): VGPR0.x of lanes 0-15 holds K=0 => q/T values
  v2f a; a.x = 0.0f; a.y = 0.0f;
  if (lo) {
    const int r = row0 + lane;
    a.x = (r < n_c) ? q_sorted[off + r] : 0.0f;
  }

  // Detached row-max shift for the 8 accumulator rows this lane sees
  float mrow[8];
  const int rbase = row0 + (lo ? 0 : 8);
#pragma unroll
  for (int r = 0; r < 8; ++r) {
    const int gr = rbase + r;
    const float qs = (gr < n_c) ? q_sorted[off + gr] : 0.0f;  // already /T
    mrow[r] = qs * (qs >= 0.0f ? kmax : kmin);
  }

  float esum[8] = {0.f, 0.f, 0.f, 0.f, 0.f, 0.f, 0.f, 0.f};

  const int ntiles = (n_c + 15) >> 4;
  for (int ct = 0; ct < ntiles; ++ct) {
    const int col0 = ct * 16;

    // B fragment (4x16 f32): VGPR0.x of lanes 0-15 holds row K=0 => k values
    v2f b; b.x = 0.0f; b.y = 0.0f;
    if (lo) {
      const int cc = col0 + lane;
      b.x = (cc < n_c) ? k_sorted[off + cc] : 0.0f;
    }

    v8f cz = {};
    // D = A x B : 16x16 tile of (q_m/T) * k_n
    v8f d = __builtin_amdgcn_wmma_f32_16x16x4_f32(
        /*neg_a=*/false, a, /*neg_b=*/false, b,
        /*c_mod=*/(short)0, cz, /*reuse_a=*/false, /*reuse_b=*/false);

    const bool colvalid = (col0 + nlane) < n_c;
    if (colvalid) {
#pragma unroll
      for (int r = 0; r < 8; ++r) {
        esum[r] += __expf(d[r] - mrow[r]);
      }
    }
  }

  // Row sums: reduce across each 16-lane half (wave32, width-16 xor shuffle)
#pragma unroll
  for (int r = 0; r < 8; ++r) {
    float v = esum[r];
    v += __shfl_xor(v, 1, 16);
    v += __shfl_xor(v, 2, 16);
    v += __shfl_xor(v, 4, 16);
    v += __shfl_xor(v, 8, 16);
    esum[r] = v;
  }

  // loss_i = (T/BASE_T) * log(pos_sum_i)  (pos_sum includes the i==j term > 0)
  if (lane == 0 || lane == 16) {
    const int base = row0 + ((lane == 16) ? 8 : 0);
#pragma unroll
    for (int r = 0; r < 8; ++r) {
      const int gr = base + r;
      if (gr < n_c) loss_row[off + gr] = LOSS_SCALE * __logf(esum[r]);
    }
  }
}

// ---------------------------------------------------------------------------
// Kernel 3: deterministic mean over the 8192 per-row losses -> scalar out.
// ---------------------------------------------------------------------------
__global__ void supclr_reduce_kernel(const float* __restrict__ loss_row,
                                     float* __restrict__ out) {
  __shared__ float s[256];
  const int t = threadIdx.x;
  float acc = 0.0f;
  for (int i = t; i < N_TOTAL; i += 256) acc += loss_row[i];
  s[t] = acc;
  __syncthreads();
  for (int stride = 128; stride > 0; stride >>= 1) {
    if (t < stride) s[t] += s[t + stride];
    __syncthreads();
  }
  if (t == 0) out[0] = s[0] * (1.0f / (float)N_TOTAL);
}

// ---------------------------------------------------------------------------
extern "C" void kernel_launch(void* const* d_in, const int* in_sizes, int n_in,
                              void* d_out, int out_size, void* d_ws, size_t ws_size,
                              hipStream_t stream) {
  (void)in_sizes; (void)n_in; (void)out_size; (void)ws_size;

  const float* q      = (const float*)d_in[0];  // feat_q, 8192 f32
  const float* k      = (const float*)d_in[1];  // feat_k, 8192 f32
  const int*   labels = (const int*)d_in[2];    // labels, 8192 int

  // Workspace carve-out (~99 KB)
  float* q_sorted = (float*)d_ws;
  float* k_sorted = q_sorted + N_TOTAL;
  float* loss_row = k_sorted + N_TOTAL;
  int*   offsets  = (int*)(loss_row + N_TOTAL);          // NCLS+1 ints
  float* kminmax  = (float*)(offsets + (NCLS + 4));      // 2 floats

  supclr_prep_kernel<<<1, 128, 0, stream>>>(q, k, labels, q_sorted, k_sorted,
                                            offsets, kminmax);

  // grid.y = 512 covers the worst case (one class holding all 8192 rows);
  // blocks past a class's row count exit immediately (uniform branch).
  dim3 grid(NCLS, N_TOTAL / 16);
  supclr_tile_kernel<<<grid, 32, 0, stream>>>(q_sorted, k_sorted, offsets,
                                              kminmax, loss_row);

  supclr_reduce_kernel<<<1, 256, 0, stream>>>(loss_row, (float*)d_out);
}